// ToyGCN_32280974197287
// MI455X (gfx1250) — compile-verified
//
#include <hip/hip_runtime.h>

typedef __attribute__((ext_vector_type(2))) float v2f;
typedef __attribute__((ext_vector_type(8))) float v8f;

#define N_NODES  100000
#define N_EDGES  3200000
#define N_GRAPHS 128
#define DIM_IN   16
#define DIM_H    64
#define DIM_C    4

// ---------------------------------------------------------------- degree
__global__ void gcn_deg_kernel(const long long* __restrict__ dst,
                               unsigned int* __restrict__ deg, int ne) {
    int e = blockIdx.x * blockDim.x + threadIdx.x;
    if (e < ne) atomicAdd(&deg[dst[e]], 1u);
}

__global__ void gcn_dinv_kernel(const unsigned int* __restrict__ deg,
                                float* __restrict__ dinv, int n) {
    int i = blockIdx.x * blockDim.x + threadIdx.x;
    if (i < n) dinv[i] = rsqrtf((float)deg[i] + 1.0f);  // +1 = self-loop
}

// ---------------------------------------------------------------- WMMA GEMM
// Hout[n x 64] = Hin[n x K] @ W[K x 64], fp32, V_WMMA_F32_16X16X4_F32.
// 256 threads = 8 waves/block; each wave: 16 rows x 64 cols (4 accumulators).
// Requires n % 16 == 0 (100000 % 16 == 0) and K % 4 == 0.
__global__ void gcn_gemm_wmma(const float* __restrict__ Hin,
                              const float* __restrict__ W,
                              float* __restrict__ Hout, int n, int K) {
    const int wave = threadIdx.x >> 5;   // 0..7
    const int lane = threadIdx.x & 31;
    const int m    = lane & 15;          // row-in-tile (A), col-in-tile (B/D)
    const int kg   = lane >> 4;          // K-pair group: 0 -> K{0,1}, 1 -> K{2,3}
    const int rowBase = blockIdx.x * 128 + wave * 16;
    if (rowBase >= n) return;            // wave-uniform: EXEC stays all-1s inside

    v8f acc0 = {}, acc1 = {}, acc2 = {}, acc3 = {};
    const float* arow = Hin + (size_t)(rowBase + m) * (size_t)K;

    for (int kt = 0; kt < K; kt += 4) {
        const int kb = kt + kg * 2;
        v2f a; a.x = arow[kb]; a.y = arow[kb + 1];
        const float* wp = W + (size_t)kb * 64 + m;
        v2f b0; b0.x = wp[0];  b0.y = wp[64 + 0];
        v2f b1; b1.x = wp[16]; b1.y = wp[64 + 16];
        v2f b2; b2.x = wp[32]; b2.y = wp[64 + 32];
        v2f b3; b3.x = wp[48]; b3.y = wp[64 + 48];
        acc0 = __builtin_amdgcn_wmma_f32_16x16x4_f32(false, a, false, b0, (short)0, acc0, false, false);
        acc1 = __builtin_amdgcn_wmma_f32_16x16x4_f32(false, a, false, b1, (short)0, acc1, false, false);
        acc2 = __builtin_amdgcn_wmma_f32_16x16x4_f32(false, a, false, b2, (short)0, acc2, false, false);
        acc3 = __builtin_amdgcn_wmma_f32_16x16x4_f32(false, a, false, b3, (short)0, acc3, false, false);
    }

    // D layout: VGPR r -> M = r + 8*kg (lanes 16-31 hold M=8..15), N = lane&15
    float* orow = Hout + (size_t)rowBase * 64;
    #pragma unroll
    for (int r = 0; r < 8; ++r) {
        const size_t rr = (size_t)(r + kg * 8) * 64;
        orow[rr +  0 + m] = acc0[r];
        orow[rr + 16 + m] = acc1[r];
        orow[rr + 32 + m] = acc2[r];
        orow[rr + 48 + m] = acc3[r];
    }
}

// ---------------------------------------------------------------- edge scatter
// 16 lanes per edge, float4 per lane: 256B coalesced gather of h[src],
// 64 global_atomic_add_f32 into acc[dst]. All traffic L2-resident (192MB L2).
__global__ void gcn_scatter_kernel(const long long* __restrict__ src,
                                   const long long* __restrict__ dst,
                                   const float* __restrict__ dinv,
                                   const float* __restrict__ h,
                                   float* __restrict__ acc, int ne) {
    int t = blockIdx.x * blockDim.x + threadIdx.x;
    int e = t >> 4;
    int sub = t & 15;
    if (e >= ne) return;
    long long s = src[e];
    long long d = dst[e];
    float w = dinv[s] * dinv[d];
    const float4* hp = (const float4*)(h + (size_t)s * 64);
    float4 v = hp[sub];
    float* ap = acc + (size_t)d * 64 + (size_t)sub * 4;
    atomicAdd(ap + 0, v.x * w);
    atomicAdd(ap + 1, v.y * w);
    atomicAdd(ap + 2, v.z * w);
    atomicAdd(ap + 3, v.w * w);
}

// ---------------------------------------------------------------- self-loop + bias + relu (in place)
__global__ void gcn_finalize_kernel(float* __restrict__ acc,
                                    const float* __restrict__ hx,
                                    const float* __restrict__ dinv,
                                    const float* __restrict__ bias, int n) {
    size_t idx = (size_t)blockIdx.x * blockDim.x + threadIdx.x;
    size_t tot = (size_t)n * 64;
    if (idx >= tot) return;
    size_t i = idx >> 6;
    int c = (int)(idx & 63);
    float di = dinv[i];
    float v = acc[idx] + di * di * hx[idx] + bias[c];
    acc[idx] = v > 0.0f ? v : 0.0f;
}

// ---------------------------------------------------------------- global add pool
__global__ void gcn_pool_kernel(const float* __restrict__ h,
                                const long long* __restrict__ batch,
                                float* __restrict__ pool, int n) {
    size_t idx = (size_t)blockIdx.x * blockDim.x + threadIdx.x;
    size_t tot = (size_t)n * 64;
    if (idx >= tot) return;
    size_t i = idx >> 6;
    int c = (int)(idx & 63);
    atomicAdd(&pool[(size_t)batch[i] * 64 + c], h[idx]);
}

// ---------------------------------------------------------------- head: [G,64]@[64,4]+bl
__global__ void gcn_head_kernel(const float* __restrict__ pool,
                                const float* __restrict__ Wl,
                                const float* __restrict__ bl,
                                float* __restrict__ out) {
    int t = blockIdx.x * blockDim.x + threadIdx.x;
    if (t >= N_GRAPHS * DIM_C) return;
    int g = t >> 2, c = t & 3;
    float s = bl[c];
    #pragma unroll 8
    for (int k = 0; k < DIM_H; ++k) s += pool[g * DIM_H + k] * Wl[k * DIM_C + c];
    out[t] = s;
}

// ----------------------------------------------------------------
extern "C" void kernel_launch(void* const* d_in, const int* in_sizes, int n_in,
                              void* d_out, int out_size, void* d_ws, size_t ws_size,
                              hipStream_t stream) {
    (void)in_sizes; (void)n_in; (void)out_size; (void)ws_size;
    const float*     x     = (const float*)d_in[0];
    const long long* ei    = (const long long*)d_in[1];   // int64 [2, E]
    const long long* batch = (const long long*)d_in[2];   // int64 [N]
    const float*     W1    = (const float*)d_in[3];
    const float*     b1    = (const float*)d_in[4];
    const float*     W2    = (const float*)d_in[5];
    const float*     b2    = (const float*)d_in[6];
    const float*     Wl    = (const float*)d_in[7];
    const float*     bl    = (const float*)d_in[8];
    const long long* src   = ei;
    const long long* dst   = ei + N_EDGES;

    // workspace layout (256B aligned), ~52 MB total
    char* ws = (char*)d_ws;
    size_t off = 0;
    auto alignUp = [](size_t v) { return (v + 255) & ~(size_t)255; };
    unsigned int* deg  = (unsigned int*)(ws + off); off = alignUp(off + (size_t)N_NODES * 4);
    float*        dinv = (float*)(ws + off);        off = alignUp(off + (size_t)N_NODES * 4);
    float*        bufA = (float*)(ws + off);        off = alignUp(off + (size_t)N_NODES * 64 * 4);
    float*        bufB = (float*)(ws + off);        off = alignUp(off + (size_t)N_NODES * 64 * 4);
    float*        pool = (float*)(ws + off);        off = alignUp(off + (size_t)N_GRAPHS * 64 * 4);

    const int B = 256;
    const int nodeElemBlocks = (int)(((size_t)N_NODES * 64 + B - 1) / B);
    const int scatterBlocks  = (int)(((size_t)N_EDGES * 16 + B - 1) / B);
    const int gemmBlocks     = (N_NODES + 127) / 128;

    // degrees -> dinv
    hipMemsetAsync(deg, 0, (size_t)N_NODES * 4, stream);
    gcn_deg_kernel<<<(N_EDGES + B - 1) / B, B, 0, stream>>>(dst, deg, N_EDGES);
    gcn_dinv_kernel<<<(N_NODES + B - 1) / B, B, 0, stream>>>(deg, dinv, N_NODES);

    // ---- layer 1: h1 = relu(Dinv A Dinv (x @ W1) + b1)
    gcn_gemm_wmma<<<gemmBlocks, B, 0, stream>>>(x, W1, bufA, N_NODES, DIM_IN);
    hipMemsetAsync(bufB, 0, (size_t)N_NODES * 64 * 4, stream);
    gcn_scatter_kernel<<<scatterBlocks, B, 0, stream>>>(src, dst, dinv, bufA, bufB, N_EDGES);
    gcn_finalize_kernel<<<nodeElemBlocks, B, 0, stream>>>(bufB, bufA, dinv, b1, N_NODES);

    // ---- layer 2: h2 = relu(Dinv A Dinv (h1 @ W2) + b2)
    gcn_gemm_wmma<<<gemmBlocks, B, 0, stream>>>(bufB, W2, bufA, N_NODES, DIM_H);
    hipMemsetAsync(bufB, 0, (size_t)N_NODES * 64 * 4, stream);
    gcn_scatter_kernel<<<scatterBlocks, B, 0, stream>>>(src, dst, dinv, bufA, bufB, N_EDGES);
    gcn_finalize_kernel<<<nodeElemBlocks, B, 0, stream>>>(bufB, bufA, dinv, b2, N_NODES);

    // ---- pool + head
    hipMemsetAsync(pool, 0, (size_t)N_GRAPHS * 64 * 4, stream);
    gcn_pool_kernel<<<nodeElemBlocks, B, 0, stream>>>(bufB, batch, pool, N_NODES);
    gcn_head_kernel<<<2, B, 0, stream>>>(pool, Wl, bl, (float*)d_out);
}